// SimpleMolecularGNN_54339926229110
// MI455X (gfx1250) — compile-verified
//
#include <hip/hip_runtime.h>

// ---------------------------------------------------------------------------
// SimpleMolecularGNN for MI455X (gfx1250)
//
// Roofline: edge scatter dominates (2 x ~1.7GB of gather+atomic traffic),
// all of which is L2-resident (192MB L2 >> 51MB feature buffers) -> atomics
// resolve in L2, HBM sees ~one pass of x/h. The GEMMs (<1.2 GFLOP total) run
// on V_WMMA_F32_16X16X4_F32 to keep exact f32 semantics at negligible cost.
// ---------------------------------------------------------------------------

typedef __attribute__((ext_vector_type(2))) float v2f;
typedef __attribute__((ext_vector_type(8))) float v8f;

// ---------------- utility fill ----------------
__global__ void fill_f32(float* __restrict__ p, long n, float v) {
  long i = (long)blockIdx.x * blockDim.x + threadIdx.x;
  if (i < n) p[i] = v;
}

// ---------------- degree / D^-1/2 ----------------
__global__ void deg_accum_kernel(const int* __restrict__ dst, float* __restrict__ deg, int E) {
  int e = blockIdx.x * blockDim.x + threadIdx.x;
  if (e < E) atomicAdd(&deg[dst[e]], 1.0f);
}

__global__ void deg_to_dinv_kernel(float* __restrict__ d, int N) {
  int n = blockIdx.x * blockDim.x + threadIdx.x;
  if (n < N) {
    float x = d[n];
    d[n] = (x > 0.0f) ? rsqrtf(x) : 0.0f;
  }
}

// ---------------- WMMA GEMM: D[M x 32] = A[M x K] @ B[K x 32] ----------------
// One wave per 16-row tile (M must be a multiple of 16). Two 16x16 N-tiles.
// f32 WMMA (16x16x4) keeps exact fp32 numerics. A-frag layout per CDNA5 ISA:
// lanes 0-15: M=l16, K={k0,k0+1}; lanes 16-31: M=l16, K={k0+2,k0+3}.
// B-frag: lanes 0-15: N=l16, K={k0,k0+1}; lanes 16-31: N=l16, K={k0+2,k0+3}.
// C/D: lanes 0-15 -> rows 0..7 (VGPR r = row r), lanes 16-31 -> rows 8..15.
template <int K>
__global__ void gemm_wmma_n32(const float* __restrict__ A, const float* __restrict__ B,
                              float* __restrict__ D, int M,
                              const float* __restrict__ bias, int do_relu) {
  int wave = blockIdx.x * (blockDim.x >> 5) + (threadIdx.x >> 5);
  int lane = threadIdx.x & 31;
  int row0 = wave * 16;
  if (row0 >= M) return;                 // wave-uniform: EXEC stays all-ones inside

  int hf  = lane >> 4;                   // 0: lanes 0-15, 1: lanes 16-31
  int l16 = lane & 15;
  const float* arp = A + (long)(row0 + l16) * K;

  v8f acc0 = {};
  v8f acc1 = {};

#pragma unroll
  for (int k0 = 0; k0 < K; k0 += 4) {
    int ka = k0 + (hf << 1);
    v2f a  = {0.0f, 0.0f};
    v2f b0 = {0.0f, 0.0f};
    v2f b1 = {0.0f, 0.0f};
    if (ka < K) {
      a.x  = arp[ka];
      b0.x = B[ka * 32 + l16];
      b1.x = B[ka * 32 + 16 + l16];
    }
    if (ka + 1 < K) {
      a.y  = arp[ka + 1];
      b0.y = B[(ka + 1) * 32 + l16];
      b1.y = B[(ka + 1) * 32 + 16 + l16];
    }
    acc0 = __builtin_amdgcn_wmma_f32_16x16x4_f32(false, a, false, b0, (short)0, acc0, false, false);
    acc1 = __builtin_amdgcn_wmma_f32_16x16x4_f32(false, a, false, b1, (short)0, acc1, false, false);
  }

  float bl = 0.0f, bh = 0.0f;
  if (bias) { bl = bias[l16]; bh = bias[16 + l16]; }

#pragma unroll
  for (int r = 0; r < 8; ++r) {
    int row  = row0 + (hf << 3) + r;
    float v0 = acc0[r] + bl;
    float v1 = acc1[r] + bh;
    if (do_relu) { v0 = fmaxf(v0, 0.0f); v1 = fmaxf(v1, 0.0f); }
    D[(long)row * 32 + l16]      = v0;
    D[(long)row * 32 + 16 + l16] = v1;
  }
}

// ---------------- edge scatter: agg[dst] += h[src] * dinv[src]*dinv[dst] ----------------
// One thread per (edge, feature). All 32 lanes of a wave share one edge ->
// coalesced 128B gather of h[src] and one-cacheline atomic burst on agg[dst];
// both buffers are L2-resident on MI455X (192MB).
__global__ void edge_scatter_kernel(const int* __restrict__ src, const int* __restrict__ dst,
                                    const float* __restrict__ dinv, const float* __restrict__ h,
                                    float* __restrict__ agg, int E) {
  long i = (long)blockIdx.x * blockDim.x + threadIdx.x;
  if (i >= (long)E * 32) return;
  int e = (int)(i >> 5);
  int f = (int)(i & 31);
  int s = src[e];
  int d = dst[e];
  float nv = dinv[s] * dinv[d];
  atomicAdd(&agg[(long)d * 32 + f], h[(long)s * 32 + f] * nv);
}

// ---------------- conv epilogue: out = agg + h*dinv^2 (self-loop) + b [, ReLU] ----------------
__global__ void finish_conv_kernel(const float* __restrict__ agg, const float* __restrict__ hin,
                                   const float* __restrict__ dinv, const float* __restrict__ bias,
                                   float* __restrict__ out, int N, int do_relu) {
  int i = blockIdx.x * blockDim.x + threadIdx.x;
  if (i >= N * 32) return;
  int n = i >> 5;
  int f = i & 31;
  float di = dinv[n];
  float v  = agg[i] + hin[i] * di * di + bias[f];
  if (do_relu) v = fmaxf(v, 0.0f);
  out[i] = v;
}

// ---------------- global mean pool ----------------
__global__ void pool_accum_kernel(const float* __restrict__ h, const int* __restrict__ batch,
                                  float* __restrict__ gsum, int N) {
  int i = blockIdx.x * blockDim.x + threadIdx.x;
  if (i >= N * 32) return;
  int n = i >> 5;
  int f = i & 31;
  atomicAdd(&gsum[batch[n] * 32 + f], h[i]);
}

__global__ void count_accum_kernel(const int* __restrict__ batch, float* __restrict__ gcnt, int N) {
  int n = blockIdx.x * blockDim.x + threadIdx.x;
  if (n < N) atomicAdd(&gcnt[batch[n]], 1.0f);
}

__global__ void pool_div_kernel(float* __restrict__ gsum, const float* __restrict__ gcnt, int G) {
  int i = blockIdx.x * blockDim.x + threadIdx.x;
  if (i >= G * 32) return;
  int g = i >> 5;
  gsum[i] = gsum[i] / fmaxf(gcnt[g], 1.0f);
}

// ---------------- final FC: out[g] = dot(fc1[g], fcW2) + fcb2 ----------------
__global__ void fc2_kernel(const float* __restrict__ a, const float* __restrict__ w,
                           const float* __restrict__ b, float* __restrict__ out, int G) {
  int g = blockIdx.x * blockDim.x + threadIdx.x;
  if (g >= G) return;
  float s = b[0];
#pragma unroll
  for (int f = 0; f < 32; ++f) s += a[(long)g * 32 + f] * w[f];
  out[g] = s;
}

// ---------------------------------------------------------------------------
extern "C" void kernel_launch(void* const* d_in, const int* in_sizes, int n_in,
                              void* d_out, int out_size, void* d_ws, size_t ws_size,
                              hipStream_t stream) {
  const float* x     = (const float*)d_in[0];
  const int*   ei    = (const int*)d_in[1];
  const int*   batch = (const int*)d_in[2];
  const float* W1    = (const float*)d_in[3];
  const float* b1    = (const float*)d_in[4];
  const float* W2    = (const float*)d_in[5];
  const float* b2    = (const float*)d_in[6];
  const float* fcW1  = (const float*)d_in[7];
  const float* fcb1  = (const float*)d_in[8];
  const float* fcW2  = (const float*)d_in[9];
  const float* fcb2  = (const float*)d_in[10];
  float*       out   = (float*)d_out;

  const int N = in_sizes[2];          // 400000 (batch array length)
  const int E = in_sizes[1] / 2;      // 6400000
  const int G = out_size;             // 20000

  const int* src = ei;                // edge_index[0]
  const int* dst = ei + E;            // edge_index[1]

  // ---- carve workspace (all buffers fully re-initialized every call) ----
  char* wsp = (char*)d_ws;
  auto carve = [&](size_t bytes) -> void* {
    void* p = (void*)wsp;
    wsp += (bytes + 255) & ~(size_t)255;
    return p;
  };
  float* dinv = (float*)carve((size_t)N * 4);        //  1.6 MB
  float* h    = (float*)carve((size_t)N * 32 * 4);   // 51.2 MB
  float* h2   = (float*)carve((size_t)N * 32 * 4);   // 51.2 MB
  float* agg  = (float*)carve((size_t)N * 32 * 4);   // 51.2 MB
  float* gsum = (float*)carve((size_t)G * 32 * 4);   //  2.6 MB
  float* gcnt = (float*)carve((size_t)G * 4);
  float* fc1  = (float*)carve((size_t)G * 32 * 4);
  (void)ws_size;

  const int BT = 256;
  auto nb = [&](long n) { return (int)((n + BT - 1) / BT); };

  // ---- degree with self-loop (init 1.0), then D^-1/2 ----
  fill_f32<<<nb(N), BT, 0, stream>>>(dinv, N, 1.0f);
  deg_accum_kernel<<<nb(E), BT, 0, stream>>>(dst, dinv, E);
  deg_to_dinv_kernel<<<nb(N), BT, 0, stream>>>(dinv, N);

  const int tilesN = N / 16;          // M multiple of 16
  const int tilesG = G / 16;

  // ---- conv1: h = x @ W1 ; agg = scatter ; h = relu(agg + selfloop + b1) ----
  gemm_wmma_n32<11><<<(tilesN + 3) / 4, 128, 0, stream>>>(x, W1, h, N, nullptr, 0);
  fill_f32<<<nb((long)N * 32), BT, 0, stream>>>(agg, (long)N * 32, 0.0f);
  edge_scatter_kernel<<<nb((long)E * 32), BT, 0, stream>>>(src, dst, dinv, h, agg, E);
  finish_conv_kernel<<<nb((long)N * 32), BT, 0, stream>>>(agg, h, dinv, b1, h, N, 1);

  // ---- conv2: h2 = h @ W2 ; scatter ; h2 = agg + selfloop + b2 ----
  gemm_wmma_n32<32><<<(tilesN + 3) / 4, 128, 0, stream>>>(h, W2, h2, N, nullptr, 0);
  fill_f32<<<nb((long)N * 32), BT, 0, stream>>>(agg, (long)N * 32, 0.0f);
  edge_scatter_kernel<<<nb((long)E * 32), BT, 0, stream>>>(src, dst, dinv, h2, agg, E);
  finish_conv_kernel<<<nb((long)N * 32), BT, 0, stream>>>(agg, h2, dinv, b2, h2, N, 0);

  // ---- global mean pool ----
  fill_f32<<<nb((long)G * 32), BT, 0, stream>>>(gsum, (long)G * 32, 0.0f);
  fill_f32<<<nb(G), BT, 0, stream>>>(gcnt, G, 0.0f);
  pool_accum_kernel<<<nb((long)N * 32), BT, 0, stream>>>(h2, batch, gsum, N);
  count_accum_kernel<<<nb(N), BT, 0, stream>>>(batch, gcnt, N);
  pool_div_kernel<<<nb((long)G * 32), BT, 0, stream>>>(gsum, gcnt, G);

  // ---- MLP head: fc1 = relu(pooled @ fcW1 + fcb1) ; out = fc1 @ fcW2 + fcb2 ----
  gemm_wmma_n32<32><<<(tilesG + 3) / 4, 128, 0, stream>>>(gsum, fcW1, fc1, G, fcb1, 1);
  fc2_kernel<<<nb(G), BT, 0, stream>>>(fc1, fcW2, fcb2, out, G);
}